// Heomd_16432544874881
// MI455X (gfx1250) — compile-verified
//
#include <hip/hip_runtime.h>
#include <math.h>

// Problem constants (from the reference): B=4, N=M=1024, D=64, fp32.
constexpr int B = 4;
constexpr int N = 1024;
constexpr int M = 1024;
constexpr int D = 64;

typedef __attribute__((ext_vector_type(2))) float v2f;
typedef __attribute__((ext_vector_type(8))) float v8f;

#define F_INF __int_as_float(0x7F800000)

// ---------------------------------------------------------------------------
// Kernel 1: row squared-norms of v1 and v2, and init colmin buffer to +inf.
// One thread per row (N == M so a single index range covers both).
// ---------------------------------------------------------------------------
__global__ void __launch_bounds__(256)
norms_init_kernel(const float* __restrict__ v1, const float* __restrict__ v2,
                  float* __restrict__ n1, float* __restrict__ n2,
                  unsigned int* __restrict__ colmin) {
  int idx = blockIdx.x * blockDim.x + threadIdx.x;
  if (idx >= B * N) return;
  const float* r1 = v1 + (size_t)idx * D;
  const float* r2 = v2 + (size_t)idx * D;
  float s1 = 0.f, s2 = 0.f;
#pragma unroll 8
  for (int d = 0; d < D; ++d) {
    s1 = fmaf(r1[d], r1[d], s1);
    s2 = fmaf(r2[d], r2[d], s2);
  }
  n1[idx] = s1;
  n2[idx] = s2;
  colmin[idx] = 0x7F800000u;  // +inf bit pattern (valid for uint atomicMin on >=0 floats)
}

// ---------------------------------------------------------------------------
// Kernel 2: main WMMA kernel.
// Each wave (wave32) owns one 16-row strip of v2 (M dim) for one batch and
// sweeps all 64 N-tiles. Gram matrix via V_WMMA_F32_16X16X4_F32, K-loop over
// D=64 in 16 steps of 4 using TWO independent accumulator chains for ILP.
// All 16 B-fragments are staged into registers first so the compiler can
// clause the loads and issue the WMMA chain without per-load waits.
// Epilogue builds squared distances from precomputed norms, tracks row mins
// in registers and pushes column mins via atomicMin on uint bit patterns
// (distances clamped >= 0 so uint order matches float order).
// ---------------------------------------------------------------------------
__global__ void __launch_bounds__(256)
chamfer_wmma_kernel(const float* __restrict__ v1, const float* __restrict__ v2,
                    const int* __restrict__ sz1, const int* __restrict__ sz2,
                    const float* __restrict__ n1, const float* __restrict__ n2,
                    unsigned int* __restrict__ colmin,
                    float* __restrict__ rowmin) {
  const int lane = threadIdx.x & 31;
  const int lo   = lane & 15;       // column / row-in-half index
  const int hi   = lane >> 4;       // which half of the wave
  const int koff = hi * 2;          // A/B fragment K offset per ISA layout

  const int wave  = (blockIdx.x * blockDim.x + threadIdx.x) >> 5;  // 0..B*M/16-1
  const int b     = wave / (M / 16);
  const int itile = wave % (M / 16);
  const int i0    = itile * 16;

  const int s1 = sz1[b];
  const int s2 = sz2[b];

  // Hoist A fragments (this wave's 16 v2 rows are reused for every N-tile).
  // A layout (16x4 f32): lane holds row (i0+lo), elements (k0+koff, k0+koff+1).
  const float* arow = v2 + ((size_t)b * M + i0 + lo) * D;
  v2f afrag[16];
#pragma unroll
  for (int kk = 0; kk < 16; ++kk) {
    afrag[kk].x = arow[kk * 4 + koff];
    afrag[kk].y = arow[kk * 4 + koff + 1];
  }

  // C/D layout: VGPR r of this lane is element (row = i0 + r + 8*hi, col = n0+lo).
  float n2row[8];
#pragma unroll
  for (int r = 0; r < 8; ++r)
    n2row[r] = n2[(size_t)b * M + i0 + r + 8 * hi];

  float rmin[8];
#pragma unroll
  for (int r = 0; r < 8; ++r) rmin[r] = F_INF;

  const float* v1b = v1 + (size_t)b * N * D;
  const float* n1b = n1 + (size_t)b * N;

  for (int nt = 0; nt < N / 16; ++nt) {
    const int n0 = nt * 16;
    const float* brow = v1b + (size_t)(n0 + lo) * D + koff;

    // Stage all B fragments first: lets the compiler emit one load clause
    // and a single s_wait before the WMMA chain instead of 16 round trips.
    v2f bfrag[16];
#pragma unroll
    for (int kk = 0; kk < 16; ++kk) {
      bfrag[kk].x = brow[kk * 4];
      bfrag[kk].y = brow[kk * 4 + 1];
    }

    // Two independent accumulator chains (even/odd K) double the matrix-pipe
    // ILP; back-to-back WMMAs writing different D registers have no hazard.
    v8f acc0 = {0.f, 0.f, 0.f, 0.f, 0.f, 0.f, 0.f, 0.f};
    v8f acc1 = {0.f, 0.f, 0.f, 0.f, 0.f, 0.f, 0.f, 0.f};
#pragma unroll
    for (int kk = 0; kk < 16; kk += 2) {
      acc0 = __builtin_amdgcn_wmma_f32_16x16x4_f32(
          false, afrag[kk],     false, bfrag[kk],     (short)0, acc0, false, false);
      acc1 = __builtin_amdgcn_wmma_f32_16x16x4_f32(
          false, afrag[kk + 1], false, bfrag[kk + 1], (short)0, acc1, false, false);
    }

    const int   j      = n0 + lo;
    const float n1col  = n1b[j];
    const bool  jvalid = (j < s1);

    float cmin = F_INF;
#pragma unroll
    for (int r = 0; r < 8; ++r) {
      const float dot = acc0[r] + acc1[r];
      float d2 = n2row[r] + n1col - 2.0f * dot;
      d2 = fmaxf(d2, 0.0f);  // kill cancellation negatives; keeps uint order valid
      // row-min over valid columns only
      rmin[r] = fminf(rmin[r], jvalid ? d2 : F_INF);
      // col-min over valid rows only
      const int irow = i0 + r + 8 * hi;
      cmin = fminf(cmin, (irow < s2) ? d2 : F_INF);
    }
    // combine the two half-wave partners covering the same column
    cmin = fminf(cmin, __shfl_xor(cmin, 16, 32));
    if (hi == 0) {
      atomicMin(&colmin[(size_t)b * N + j], __float_as_uint(cmin));
    }
  }

  // Row-min reduction across the 16 lanes of each half (xor 1,2,4,8 stays
  // inside a 16-lane group), then one lane per half writes its 8 rows.
#pragma unroll
  for (int r = 0; r < 8; ++r) {
    float v = rmin[r];
    v = fminf(v, __shfl_xor(v, 1, 32));
    v = fminf(v, __shfl_xor(v, 2, 32));
    v = fminf(v, __shfl_xor(v, 4, 32));
    v = fminf(v, __shfl_xor(v, 8, 32));
    if (lo == 0) {
      rowmin[(size_t)b * M + i0 + r + 8 * hi] = v;
    }
  }
}

// ---------------------------------------------------------------------------
// Kernel 3: per-batch final reduction. sqrt the squared mins, sum valid
// entries, normalize by (sz1 + sz2).
// ---------------------------------------------------------------------------
__global__ void __launch_bounds__(256)
reduce_kernel(const float* __restrict__ rowmin,
              const unsigned int* __restrict__ colmin,
              const int* __restrict__ sz1, const int* __restrict__ sz2,
              float* __restrict__ out) {
  const int b = blockIdx.x;
  const int t = threadIdx.x;
  const int s1 = sz1[b];
  const int s2 = sz2[b];

  float sum = 0.f;
  for (int i = t; i < M; i += 256)
    if (i < s2) sum += sqrtf(rowmin[(size_t)b * M + i]);
  for (int j = t; j < N; j += 256)
    if (j < s1) sum += sqrtf(__uint_as_float(colmin[(size_t)b * N + j]));

  __shared__ float red[256];
  red[t] = sum;
  __syncthreads();
  for (int s = 128; s > 0; s >>= 1) {
    if (t < s) red[t] += red[t + s];
    __syncthreads();
  }
  if (t == 0) out[b] = red[0] / (float)(s1 + s2);
}

// ---------------------------------------------------------------------------
// Launch: inputs are (v1, am1, sz1, v2, am2, sz2); am1/am2 unused.
// Workspace layout (floats): n1[B*N] | n2[B*M] | colmin[B*N] | rowmin[B*M]
// = 64 KB total.
// ---------------------------------------------------------------------------
extern "C" void kernel_launch(void* const* d_in, const int* in_sizes, int n_in,
                              void* d_out, int out_size, void* d_ws, size_t ws_size,
                              hipStream_t stream) {
  const float* v1  = (const float*)d_in[0];
  const int*   sz1 = (const int*)d_in[2];
  const float* v2  = (const float*)d_in[3];
  const int*   sz2 = (const int*)d_in[5];
  float* out = (float*)d_out;

  float*        n1     = (float*)d_ws;
  float*        n2     = n1 + B * N;
  unsigned int* colmin = (unsigned int*)(n2 + B * M);
  float*        rowmin = (float*)(colmin + B * N);

  norms_init_kernel<<<(B * N + 255) / 256, 256, 0, stream>>>(v1, v2, n1, n2, colmin);

  // B * (M/16) waves = 256 waves; 8 waves (256 threads) per block -> 32 blocks.
  const int total_waves = B * (M / 16);
  chamfer_wmma_kernel<<<(total_waves * 32) / 256, 256, 0, stream>>>(
      v1, v2, sz1, sz2, n1, n2, colmin, rowmin);

  reduce_kernel<<<B, 256, 0, stream>>>(rowmin, colmin, sz1, sz2, out);
}